// eSEN_Block_78176994721864
// MI455X (gfx1250) — compile-verified
//
#include <hip/hip_runtime.h>
#include <math.h>

// ---------------------------------------------------------------------------
// eSEN block for MI455X (gfx1250, wave32). All big GEMMs use
// V_WMMA_F32_16X16X4_F32 (exact fp32 math). Tile fills use
// GLOBAL_LOAD_ASYNC_TO_LDS_B32 (ASYNCcnt) with double-buffered LDS so the
// async fills overlap the WMMA stream.
// ---------------------------------------------------------------------------

typedef float v2f __attribute__((ext_vector_type(2)));
typedef float v8f __attribute__((ext_vector_type(8)));

#define NE 40000      // edges
#define NN 2000       // nodes
#define LL 9          // (LMAX+1)^2
#define CCH 128       // channels

__device__ __forceinline__ v8f wmma_f32(v2f a, v2f b, v8f c) {
  // 8 args: (neg_a, A, neg_b, B, c_mod, C, reuse_a, reuse_b)
  return __builtin_amdgcn_wmma_f32_16x16x4_f32(false, a, false, b, (short)0, c,
                                               false, false);
}

// Async global -> LDS copy (4 bytes). lds_off = low 32 bits of generic addr
// of the shared destination (LDS aperture sits in addr[63:32]).
__device__ __forceinline__ void async_ld32(const float* lds_dst,
                                           const float* gsrc) {
  unsigned int loff = (unsigned int)(unsigned long long)lds_dst;
  asm volatile("global_load_async_to_lds_b32 %0, %1, off"
               :
               : "v"(loff), "v"(gsrc)
               : "memory");
}
__device__ __forceinline__ void wait_async() {
  asm volatile("s_wait_asynccnt 0x0" ::: "memory");
}

__device__ __forceinline__ float block_sum(float v, float* buf, int t, int n) {
  buf[t] = v;
  __syncthreads();
  for (int s = n >> 1; s > 0; s >>= 1) {
    if (t < s) buf[t] += buf[t + s];
    __syncthreads();
  }
  float r = buf[0];
  __syncthreads();
  return r;
}

__device__ __forceinline__ float sigmoidf(float x) {
  return 1.f / (1.f + __expf(-x));
}
__device__ __forceinline__ float siluf(float x) { return x * sigmoidf(x); }

static __device__ const int   DEGD[9]  = {0, 1, 1, 1, 2, 2, 2, 2, 2};
static __device__ const float BALW[9]  = {1.f / 3,  1.f / 9,  1.f / 9,
                                          1.f / 9,  1.f / 15, 1.f / 15,
                                          1.f / 15, 1.f / 15, 1.f / 15};
static __device__ const int   MPERM[9]   = {0, 2, 6, 3, 7, 1, 5, 8, 4};
static __device__ const int   RADBASE[9] = {0, 256, 512, 768, 1024, 768, 1024, 1280, 1280};

// ---------------------------------------------------------------------------
// K1: rms_norm_sh  (one block of 128 threads per node)
// ---------------------------------------------------------------------------
__global__ void rms_norm_sh_kernel(const float* __restrict__ x,
                                   const float* __restrict__ w,   // (3,128)
                                   const float* __restrict__ b,   // (128)
                                   float* __restrict__ out) {
  __shared__ float red[CCH];
  int n = blockIdx.x, c = threadIdx.x;
  const float* xp = x + (size_t)n * (LL * CCH);
  float v[LL];
#pragma unroll
  for (int l = 0; l < LL; ++l) v[l] = xp[l * CCH + c];
  float mu0 = block_sum(v[0], red, c, CCH) * (1.f / CCH);
  v[0] -= mu0;
  float s = 0.f;
#pragma unroll
  for (int l = 0; l < LL; ++l) s += BALW[l] * v[l] * v[l];
  float fn = block_sum(s, red, c, CCH) * (1.f / CCH);
  float sc = rsqrtf(fn + 1e-12f);
  float* op = out + (size_t)n * (LL * CCH);
#pragma unroll
  for (int l = 0; l < LL; ++l) {
    float o = v[l] * sc * w[DEGD[l] * CCH + c];
    if (l == 0) o += b[c];
    op[l * CCH + c] = o;
  }
}

// ---------------------------------------------------------------------------
// K2: radial MLP layers 0,1 (64 threads per edge)
// ---------------------------------------------------------------------------
__global__ void radial_mlp01_kernel(
    const float* __restrict__ xe, const float* __restrict__ rw0,
    const float* __restrict__ rb0, const float* __restrict__ g0,
    const float* __restrict__ bb0, const float* __restrict__ rw1,
    const float* __restrict__ rb1, const float* __restrict__ g1,
    const float* __restrict__ bb1, float* __restrict__ outh) {
  __shared__ float xin[128];
  __shared__ float hb[64];
  __shared__ float red[64];
  int e = blockIdx.x, t = threadIdx.x;
  async_ld32(&xin[t], xe + (size_t)e * 128 + t);
  async_ld32(&xin[t + 64], xe + (size_t)e * 128 + 64 + t);
  wait_async();
  __syncthreads();
  float a = rb0[t];
  const float* wr = rw0 + t * 128;
  for (int k = 0; k < 128; ++k) a += wr[k] * xin[k];
  float mu = block_sum(a, red, t, 64) * (1.f / 64);
  float d = a - mu;
  float var = block_sum(d * d, red, t, 64) * (1.f / 64);
  float xh = d * rsqrtf(var + 1e-5f) * g0[t] + bb0[t];
  hb[t] = siluf(xh);
  __syncthreads();
  float a1 = rb1[t];
  const float* wr1 = rw1 + t * 64;
  for (int k = 0; k < 64; ++k) a1 += wr1[k] * hb[k];
  mu = block_sum(a1, red, t, 64) * (1.f / 64);
  d = a1 - mu;
  var = block_sum(d * d, red, t, 64) * (1.f / 64);
  xh = d * rsqrtf(var + 1e-5f) * g1[t] + bb1[t];
  outh[(size_t)e * 64 + t] = siluf(xh);
}

// ---------------------------------------------------------------------------
// K3: WMMA fp32 GEMM: C(M,N) = act(A(M,K)@B(N,K)^T + bias)
// 256 threads = 8 waves (2x4), wave tile 32x32 (2x2 regs), block tile 64x128.
// Double-buffered async-to-LDS tile fills overlap the WMMA stream.
// ---------------------------------------------------------------------------
enum { EPI_NONE = 0, EPI_SILU = 1, EPI_ADD = 2 };

template <int EPI>
__global__ void wmma_gemm_kernel(const float* __restrict__ A, int lda,
                                 const float* __restrict__ B, int ldb,
                                 const float* __restrict__ bias,
                                 float* __restrict__ C, int ldc, int M, int K) {
  __shared__ float As[2][64][17];
  __shared__ float Bs[2][128][17];
  int tid = threadIdx.x;
  int wave = tid >> 5, lane = tid & 31;
  int wm = wave & 1, wn = wave >> 1;
  int m0 = blockIdx.x * 64;
  int n0 = blockIdx.y * 128;
  v8f a00 = {0.f, 0.f, 0.f, 0.f, 0.f, 0.f, 0.f, 0.f};
  v8f a01 = a00, a10 = a00, a11 = a00;
  int lr = lane & 15;
  int kh = (lane >> 4) << 1;  // 0 or 2
  int ar = wm * 32 + lr;      // A tile row (mi=0); +16 for mi=1
  int br = wn * 32 + lr;      // B tile row (nj=0); +16 for nj=1

  auto load_tile = [&](int buf, int k0) {
#pragma unroll
    for (int i = 0; i < 4; ++i) {  // A tile 64x16
      int t = tid + i * 256;
      int r = t >> 4, cx = t & 15;
      int gm = m0 + r;
      if (gm < M)
        async_ld32(&As[buf][r][cx], A + (size_t)gm * lda + k0 + cx);
      else
        As[buf][r][cx] = 0.f;
    }
#pragma unroll
    for (int i = 0; i < 8; ++i) {  // B tile 128x16
      int t = tid + i * 256;
      int r = t >> 4, cx = t & 15;
      async_ld32(&Bs[buf][r][cx], B + (size_t)(n0 + r) * ldb + k0 + cx);
    }
  };

  int nk = K >> 4;
  load_tile(0, 0);
  int cur = 0;
  for (int ki = 0; ki < nk; ++ki) {
    wait_async();       // this wave's fills for tile ki done
    __syncthreads();    // everyone filled / everyone done with other buffer
    if (ki + 1 < nk) load_tile(cur ^ 1, (ki + 1) << 4);
#pragma unroll
    for (int kk = 0; kk < 16; kk += 4) {
      v2f fa0, fa1, fb0, fb1;
      fa0.x = As[cur][ar][kk + kh];
      fa0.y = As[cur][ar][kk + kh + 1];
      fa1.x = As[cur][ar + 16][kk + kh];
      fa1.y = As[cur][ar + 16][kk + kh + 1];
      fb0.x = Bs[cur][br][kk + kh];
      fb0.y = Bs[cur][br][kk + kh + 1];
      fb1.x = Bs[cur][br + 16][kk + kh];
      fb1.y = Bs[cur][br + 16][kk + kh + 1];
      a00 = wmma_f32(fa0, fb0, a00);
      a01 = wmma_f32(fa0, fb1, a01);
      a10 = wmma_f32(fa1, fb0, a10);
      a11 = wmma_f32(fa1, fb1, a11);
    }
    cur ^= 1;
  }
  int n = n0 + wn * 32 + lr;
  int mb = m0 + wm * 32 + ((lane >> 4) << 3);
  float bv0 = bias ? bias[n] : 0.f;
  float bv1 = bias ? bias[n + 16] : 0.f;
#pragma unroll
  for (int r = 0; r < 8; ++r) {
    int m = mb + r;
    if (m < M) {
      float v0 = a00[r] + bv0, v1 = a01[r] + bv1;
      if (EPI == EPI_SILU) { v0 = siluf(v0); v1 = siluf(v1); }
      if (EPI == EPI_ADD) {
        C[(size_t)m * ldc + n] += v0;
        C[(size_t)m * ldc + n + 16] += v1;
      } else {
        C[(size_t)m * ldc + n] = v0;
        C[(size_t)m * ldc + n + 16] = v1;
      }
    }
    int m2 = mb + 16 + r;
    if (m2 < M) {
      float v0 = a10[r] + bv0, v1 = a11[r] + bv1;
      if (EPI == EPI_SILU) { v0 = siluf(v0); v1 = siluf(v1); }
      if (EPI == EPI_ADD) {
        C[(size_t)m2 * ldc + n] += v0;
        C[(size_t)m2 * ldc + n + 16] += v1;
      } else {
        C[(size_t)m2 * ldc + n] = v0;
        C[(size_t)m2 * ldc + n + 16] = v1;
      }
    }
  }
}

// ---------------------------------------------------------------------------
// K4: complex WMMA GEMM for SO(2) m>0 blocks:
//   Yr = Xp@Wr^T - Xn@Wi^T ;  Yi = Xn@Wr^T + Xp@Wi^T
// Same 64x128 block tile, 2x2 wave tiling, double-buffered async fills.
// ---------------------------------------------------------------------------
__global__ void wmma_cgemm_kernel(const float* __restrict__ Xp,
                                  const float* __restrict__ Xn, int lda,
                                  const float* __restrict__ Wr,
                                  const float* __restrict__ Wi,
                                  float* __restrict__ Yr, float* __restrict__ Yi,
                                  int ldc, int M, int K) {
  __shared__ float Aps[2][64][17];
  __shared__ float Ans[2][64][17];
  __shared__ float Brs[2][128][17];
  __shared__ float Bis[2][128][17];
  int tid = threadIdx.x;
  int wave = tid >> 5, lane = tid & 31;
  int wm = wave & 1, wn = wave >> 1;
  int m0 = blockIdx.x * 64, n0 = blockIdx.y * 128;
  v8f r00 = {0.f, 0.f, 0.f, 0.f, 0.f, 0.f, 0.f, 0.f};
  v8f r01 = r00, r10 = r00, r11 = r00;
  v8f i00 = r00, i01 = r00, i10 = r00, i11 = r00;
  int lr = lane & 15;
  int kh = (lane >> 4) << 1;
  int ar = wm * 32 + lr;
  int br = wn * 32 + lr;

  auto load_tile = [&](int buf, int k0) {
#pragma unroll
    for (int i = 0; i < 4; ++i) {
      int t = tid + i * 256;
      int r = t >> 4, cx = t & 15;
      int gm = m0 + r;
      if (gm < M) {
        async_ld32(&Aps[buf][r][cx], Xp + (size_t)gm * lda + k0 + cx);
        async_ld32(&Ans[buf][r][cx], Xn + (size_t)gm * lda + k0 + cx);
      } else {
        Aps[buf][r][cx] = 0.f;
        Ans[buf][r][cx] = 0.f;
      }
    }
#pragma unroll
    for (int i = 0; i < 8; ++i) {
      int t = tid + i * 256;
      int r = t >> 4, cx = t & 15;
      async_ld32(&Brs[buf][r][cx], Wr + (size_t)(n0 + r) * K + k0 + cx);
      async_ld32(&Bis[buf][r][cx], Wi + (size_t)(n0 + r) * K + k0 + cx);
    }
  };

  int nk = K >> 4;
  load_tile(0, 0);
  int cur = 0;
  for (int ki = 0; ki < nk; ++ki) {
    wait_async();
    __syncthreads();
    if (ki + 1 < nk) load_tile(cur ^ 1, (ki + 1) << 4);
#pragma unroll
    for (int kk = 0; kk < 16; kk += 4) {
      v2f ap0, ap1, an0, an1, nn0, nn1, wr0, wr1, wi0, wi1;
      ap0.x = Aps[cur][ar][kk + kh];      ap0.y = Aps[cur][ar][kk + kh + 1];
      ap1.x = Aps[cur][ar + 16][kk + kh]; ap1.y = Aps[cur][ar + 16][kk + kh + 1];
      an0.x = Ans[cur][ar][kk + kh];      an0.y = Ans[cur][ar][kk + kh + 1];
      an1.x = Ans[cur][ar + 16][kk + kh]; an1.y = Ans[cur][ar + 16][kk + kh + 1];
      wr0.x = Brs[cur][br][kk + kh];      wr0.y = Brs[cur][br][kk + kh + 1];
      wr1.x = Brs[cur][br + 16][kk + kh]; wr1.y = Brs[cur][br + 16][kk + kh + 1];
      wi0.x = Bis[cur][br][kk + kh];      wi0.y = Bis[cur][br][kk + kh + 1];
      wi1.x = Bis[cur][br + 16][kk + kh]; wi1.y = Bis[cur][br + 16][kk + kh + 1];
      nn0.x = -an0.x; nn0.y = -an0.y;
      nn1.x = -an1.x; nn1.y = -an1.y;
      r00 = wmma_f32(ap0, wr0, r00);  r00 = wmma_f32(nn0, wi0, r00);
      r01 = wmma_f32(ap0, wr1, r01);  r01 = wmma_f32(nn0, wi1, r01);
      r10 = wmma_f32(ap1, wr0, r10);  r10 = wmma_f32(nn1, wi0, r10);
      r11 = wmma_f32(ap1, wr1, r11);  r11 = wmma_f32(nn1, wi1, r11);
      i00 = wmma_f32(an0, wr0, i00);  i00 = wmma_f32(ap0, wi0, i00);
      i01 = wmma_f32(an0, wr1, i01);  i01 = wmma_f32(ap0, wi1, i01);
      i10 = wmma_f32(an1, wr0, i10);  i10 = wmma_f32(ap1, wi0, i10);
      i11 = wmma_f32(an1, wr1, i11);  i11 = wmma_f32(ap1, wi1, i11);
    }
    cur ^= 1;
  }
  int n = n0 + wn * 32 + lr;
  int mb = m0 + wm * 32 + ((lane >> 4) << 3);
#pragma unroll
  for (int r = 0; r < 8; ++r) {
    int m = mb + r;
    if (m < M) {
      Yr[(size_t)m * ldc + n] = r00[r];
      Yr[(size_t)m * ldc + n + 16] = r01[r];
      Yi[(size_t)m * ldc + n] = i00[r];
      Yi[(size_t)m * ldc + n + 16] = i01[r];
    }
    int m2 = mb + 16 + r;
    if (m2 < M) {
      Yr[(size_t)m2 * ldc + n] = r10[r];
      Yr[(size_t)m2 * ldc + n + 16] = r11[r];
      Yi[(size_t)m2 * ldc + n] = i10[r];
      Yi[(size_t)m2 * ldc + n + 16] = i11[r];
    }
  }
}

// ---------------------------------------------------------------------------
// K5: gather h[src]/h[dst] (async to LDS), Wigner-rotate, M-permute,
// radial-scale -> xm1.  256 threads per edge.
// ---------------------------------------------------------------------------
__global__ void build_xm1_kernel(const float* __restrict__ h,
                                 const int* __restrict__ ei,
                                 const float* __restrict__ wigner,
                                 const float* __restrict__ rad,
                                 float* __restrict__ xm1) {
  __shared__ float hs[9 * 128];
  __shared__ float hd[9 * 128];
  __shared__ float wg[81];
  int e = blockIdx.x, t = threadIdx.x;
  int src = ei[e], dst = ei[NE + e];
  const float* hsp = h + (size_t)src * 1152;
  const float* hdp = h + (size_t)dst * 1152;
#pragma unroll
  for (int i = 0; i < 5; ++i) {
    int j = t + i * 256;
    if (j < 1152) {
      async_ld32(&hs[j], hsp + j);
      async_ld32(&hd[j], hdp + j);
    }
  }
  if (t < 81) wg[t] = wigner[(size_t)e * 81 + t];
  wait_async();
  __syncthreads();
  int c = t;  // concat channel 0..255
  int cc = c & 127;
  const float* srcb = (c < 128) ? hs : hd;
  float v[9];
#pragma unroll
  for (int i = 0; i < 9; ++i) v[i] = srcb[i * 128 + cc];
  const float* re = rad + (size_t)e * 1536;
  float* xo = xm1 + (size_t)e * 2304;
#pragma unroll
  for (int j = 0; j < 9; ++j) {
    const float* wr = wg + MPERM[j] * 9;
    float acc = 0.f;
#pragma unroll
    for (int k = 0; k < 9; ++k) acc += wr[k] * v[k];
    xo[j * 256 + c] = acc * re[RADBASE[j] + c];
  }
}

// ---------------------------------------------------------------------------
// K6: gate conv1 output + build conv2 m-permuted input xm2 (128 threads/edge)
// ---------------------------------------------------------------------------
__global__ void gate1_kernel(const float* __restrict__ y0full,
                             const float* __restrict__ yr1,
                             const float* __restrict__ yi1,
                             const float* __restrict__ yr2,
                             const float* __restrict__ yi2,
                             float* __restrict__ xm2) {
  int e = blockIdx.x, c = threadIdx.x;
  const float* y0 = y0full + (size_t)e * 640;
  float g0 = sigmoidf(y0[c]);        // degree-1 gate
  float g1 = sigmoidf(y0[128 + c]);  // degree-2 gate
  float* xo = xm2 + (size_t)e * 1152;
  xo[c] = siluf(y0[256 + c]);                              // j0 (l=0)
  xo[128 + c] = y0[384 + c] * g0;                          // j1 (deg1)
  xo[256 + c] = y0[512 + c] * g1;                          // j2 (deg2)
  xo[384 + c] = yr1[(size_t)e * 256 + c] * g0;             // j3
  xo[512 + c] = yr1[(size_t)e * 256 + 128 + c] * g1;       // j4
  xo[640 + c] = yi1[(size_t)e * 256 + c] * g0;             // j5
  xo[768 + c] = yi1[(size_t)e * 256 + 128 + c] * g1;       // j6
  xo[896 + c] = yr2[(size_t)e * 128 + c] * g1;             // j7
  xo[1024 + c] = yi2[(size_t)e * 128 + c] * g1;            // j8
}

// ---------------------------------------------------------------------------
// K7: inverse-perm + envelope + wigner_inv rotate + atomic scatter-add
// ---------------------------------------------------------------------------
__global__ void scatter_kernel(const float* __restrict__ y0b,
                               const float* __restrict__ yr1b,
                               const float* __restrict__ yi1b,
                               const float* __restrict__ yr2b,
                               const float* __restrict__ yi2b,
                               const float* __restrict__ wigner_inv,
                               const float* __restrict__ dist,
                               const int* __restrict__ ei,
                               float* __restrict__ out) {
  __shared__ float wg[81];
  int e = blockIdx.x, c = threadIdx.x;
  if (c < 81) wg[c] = wigner_inv[(size_t)e * 81 + c];
  __syncthreads();
  float dd = dist[e] * (1.f / 6.f);
  float d2 = dd * dd;
  float d5 = d2 * d2 * dd;
  float env = (dd < 1.f) ? (1.f - 21.f * d5 + 35.f * d5 * dd - 15.f * d5 * d2)
                         : 0.f;
  float v[9];  // conv2 output rows in ORIGINAL l-order
  v[0] = y0b[(size_t)e * 384 + c];
  v[1] = yi1b[(size_t)e * 256 + c];
  v[2] = y0b[(size_t)e * 384 + 128 + c];
  v[3] = yr1b[(size_t)e * 256 + c];
  v[4] = yi2b[(size_t)e * 128 + c];
  v[5] = yi1b[(size_t)e * 256 + 128 + c];
  v[6] = y0b[(size_t)e * 384 + 256 + c];
  v[7] = yr1b[(size_t)e * 256 + 128 + c];
  v[8] = yr2b[(size_t)e * 128 + c];
  int dst = ei[NE + e];
  float* op = out + (size_t)dst * 1152 + c;
#pragma unroll
  for (int r = 0; r < 9; ++r) {
    const float* wr = wg + r * 9;
    float acc = 0.f;
#pragma unroll
    for (int i = 0; i < 9; ++i) acc += wr[i] * v[i];
    atomicAdd(op + r * 128, acc * env);
  }
}

// ---------------------------------------------------------------------------
// K8: gate for second block (in-place on h3), 128 threads per node
// ---------------------------------------------------------------------------
__global__ void gate2_kernel(const float* __restrict__ gs,
                             float* __restrict__ h3) {
  int n = blockIdx.x, c = threadIdx.x;
  float g0 = sigmoidf(gs[(size_t)n * 256 + c]);
  float g1 = sigmoidf(gs[(size_t)n * 256 + 128 + c]);
  float* hp = h3 + (size_t)n * 1152;
  hp[c] = siluf(hp[c]);
#pragma unroll
  for (int l = 1; l < 9; ++l) hp[l * 128 + c] *= (l < 4) ? g0 : g1;
}

// ---------------------------------------------------------------------------
extern "C" void kernel_launch(void* const* d_in, const int* in_sizes, int n_in,
                              void* d_out, int out_size, void* d_ws,
                              size_t ws_size, hipStream_t stream) {
  (void)in_sizes; (void)n_in; (void)out_size; (void)ws_size;
  const float* x       = (const float*)d_in[0];
  const float* x_edge  = (const float*)d_in[1];
  const float* e_dist  = (const float*)d_in[2];
  const int*   e_index = (const int*)d_in[3];
  const float* wigner  = (const float*)d_in[4];
  const float* wig_inv = (const float*)d_in[5];
  const float* n1_w = (const float*)d_in[6];
  const float* n1_b = (const float*)d_in[7];
  const float* n2_w = (const float*)d_in[8];
  const float* n2_b = (const float*)d_in[9];
  const float* rw0  = (const float*)d_in[10];
  const float* rb0  = (const float*)d_in[11];
  const float* ln0g = (const float*)d_in[12];
  const float* ln0b = (const float*)d_in[13];
  const float* rw1  = (const float*)d_in[14];
  const float* rb1  = (const float*)d_in[15];
  const float* ln1g = (const float*)d_in[16];
  const float* ln1b = (const float*)d_in[17];
  const float* rw2  = (const float*)d_in[18];
  const float* rb2  = (const float*)d_in[19];
  const float* c1W0 = (const float*)d_in[20];
  const float* c1b0 = (const float*)d_in[21];
  const float* c1W1 = (const float*)d_in[22];
  const float* c1W2 = (const float*)d_in[23];
  const float* c2W0 = (const float*)d_in[24];
  const float* c2b0 = (const float*)d_in[25];
  const float* c2W1 = (const float*)d_in[26];
  const float* c2W2 = (const float*)d_in[27];
  const float* smW  = (const float*)d_in[28];
  const float* smb  = (const float*)d_in[29];
  const float* s1W  = (const float*)d_in[30];
  const float* s1b  = (const float*)d_in[31];
  const float* s2W  = (const float*)d_in[32];
  const float* s2b  = (const float*)d_in[33];

  float* out = (float*)d_out;
  float* ws  = (float*)d_ws;

  // ---- workspace layout (floats), phase-aliased -------------------------
  float* h    = ws;                    //  2,304,000
  float* radh = h + 2304000;           //  2,560,000
  float* rad  = radh + 2560000;        // 61,440,000
  float* xm1  = rad + 61440000;        // 92,160,000  (peak)
  // conv1 outputs alias the (dead) rad region:
  float* y0full = rad;                 // 25,600,000
  float* yr1 = y0full + 25600000;      // 10,240,000
  float* yi1 = yr1 + 10240000;         // 10,240,000
  float* yr2 = yi1 + 10240000;         //  5,120,000
  float* yi2 = yr2 + 5120000;          //  5,120,000
  // conv2 input aliases first half of (dead) xm1, outputs second half:
  float* xm2  = xm1;                   // 46,080,000
  float* y0b  = xm1 + 46080000;        // 15,360,000
  float* yr1b = y0b + 15360000;        // 10,240,000
  float* yi1b = yr1b + 10240000;       // 10,240,000
  float* yr2b = yi1b + 10240000;       //  5,120,000
  float* yi2b = yr2b + 5120000;        //  5,120,000
  // node-block temps alias dead radial buffers:
  float* h2 = radh;                    //  2,304,000
  float* gsb = rad;                    //    512,000
  float* h3 = rad + 512000;            //  2,304,000

  // out = x  (residual base for edge scatter)
  hipMemcpyAsync(out, x, (size_t)NN * LL * CCH * sizeof(float),
                 hipMemcpyDeviceToDevice, stream);

  // ---- block 1: norm + radial -----------------------------------------
  rms_norm_sh_kernel<<<NN, 128, 0, stream>>>(x, n1_w, n1_b, h);
  radial_mlp01_kernel<<<NE, 64, 0, stream>>>(x_edge, rw0, rb0, ln0g, ln0b,
                                             rw1, rb1, ln1g, ln1b, radh);
  // rad = radh @ rw2^T + rb2   (40000x64 @ 64x1536)
  wmma_gemm_kernel<EPI_NONE><<<dim3(625, 12), 256, 0, stream>>>(
      radh, 64, rw2, 64, rb2, rad, 1536, NE, 64);

  // ---- message build ---------------------------------------------------
  build_xm1_kernel<<<NE, 256, 0, stream>>>(h, e_index, wigner, rad, xm1);

  // ---- SO(2) conv 1 ----------------------------------------------------
  wmma_gemm_kernel<EPI_NONE><<<dim3(625, 5), 256, 0, stream>>>(
      xm1, 2304, c1W0, 768, c1b0, y0full, 640, NE, 768);           // m=0 (+gates)
  wmma_cgemm_kernel<<<dim3(625, 2), 256, 0, stream>>>(
      xm1 + 768, xm1 + 1280, 2304, c1W1, c1W1 + 256 * 512, yr1, yi1, 256, NE,
      512);                                                        // m=1
  wmma_cgemm_kernel<<<dim3(625, 1), 256, 0, stream>>>(
      xm1 + 1792, xm1 + 2048, 2304, c1W2, c1W2 + 128 * 256, yr2, yi2, 128, NE,
      256);                                                        // m=2

  // ---- gate + re-permute into conv2 input ------------------------------
  gate1_kernel<<<NE, 128, 0, stream>>>(y0full, yr1, yi1, yr2, yi2, xm2);

  // ---- SO(2) conv 2 ----------------------------------------------------
  wmma_gemm_kernel<EPI_NONE><<<dim3(625, 3), 256, 0, stream>>>(
      xm2, 1152, c2W0, 384, c2b0, y0b, 384, NE, 384);              // m=0
  wmma_cgemm_kernel<<<dim3(625, 2), 256, 0, stream>>>(
      xm2 + 384, xm2 + 640, 1152, c2W1, c2W1 + 256 * 256, yr1b, yi1b, 256, NE,
      256);                                                        // m=1
  wmma_cgemm_kernel<<<dim3(625, 1), 256, 0, stream>>>(
      xm2 + 896, xm2 + 1024, 1152, c2W2, c2W2 + 128 * 128, yr2b, yi2b, 128, NE,
      128);                                                        // m=2

  // ---- envelope + inverse Wigner + scatter-add into out ----------------
  scatter_kernel<<<NE, 128, 0, stream>>>(y0b, yr1b, yi1b, yr2b, yi2b, wig_inv,
                                         e_dist, e_index, out);

  // ---- block 2: node feed-forward --------------------------------------
  rms_norm_sh_kernel<<<NN, 128, 0, stream>>>(out, n2_w, n2_b, h2);
  // gs = silu(h2[:,0,:] @ smW^T + smb)   (2000x128 @ 128x256)
  wmma_gemm_kernel<EPI_SILU><<<dim3(32, 2), 256, 0, stream>>>(
      h2, 1152, smW, 128, smb, gsb, 256, NN, 128);
  // s1: per-l degree-selected 128x128 GEMMs
  static const int hDEG[9] = {0, 1, 1, 1, 2, 2, 2, 2, 2};
  for (int l = 0; l < 9; ++l) {
    wmma_gemm_kernel<EPI_NONE><<<dim3(32, 1), 256, 0, stream>>>(
        h2 + l * 128, 1152, s1W + hDEG[l] * 128 * 128, 128,
        (l == 0) ? s1b : (const float*)nullptr, h3 + l * 128, 1152, NN, 128);
  }
  gate2_kernel<<<NN, 128, 0, stream>>>(gsb, h3);
  // s2: accumulate into out (residual)
  for (int l = 0; l < 9; ++l) {
    wmma_gemm_kernel<EPI_ADD><<<dim3(32, 1), 256, 0, stream>>>(
        h3 + l * 128, 1152, s2W + hDEG[l] * 128 * 128, 128,
        (l == 0) ? s2b : (const float*)nullptr, out + l * 128, 1152, NN, 128);
  }
}